// MyUnidirectionalRNN_70566312673620
// MI455X (gfx1250) — compile-verified
//
#include <hip/hip_runtime.h>
#include <hip/hip_bf16.h>

// ---------------------------------------------------------------------------
// MyUnidirectionalRNN on MI455X (gfx1250), f32 end-to-end.
//   Phase 1: wx[m,n] = sum_k x[m,k]*W[n,k] + Ub[n] + b[n]   (WMMA f32 GEMM,
//            TDM tensor_load_to_lds double-buffered tiles)
//   Phase 2: state[t] = wx[t] + state[t-1] @ Uw^T           (WMMA f32 scan,
//            Uw slice TDM-loaded into LDS once, resident for all 2048 steps)
// state buffer == d_out (in-place: wx[t] is overwritten by state[t]).
// ---------------------------------------------------------------------------

typedef __attribute__((ext_vector_type(2))) float v2f;
typedef __attribute__((ext_vector_type(8))) float v8f;
typedef __attribute__((ext_vector_type(4))) unsigned int u32x4;
typedef __attribute__((ext_vector_type(8))) int i32x8;
typedef __attribute__((ext_vector_type(4))) int i32x4;

#define BATCH 64
#define SEQ   2048
#define KIN   512
#define HH    1024
#define MTOT  (BATCH * SEQ)            // 131072 rows of the big GEMM
#define THH   ((size_t)SEQ * HH)       // per-batch-row stride in d_out

// Raw LDS byte offset of a generic pointer into __shared__ (addrspacecast).
typedef __attribute__((address_space(3))) char lds_byte;
static __device__ __forceinline__ unsigned lds_off(const void* p)
{
    return (unsigned)(unsigned long long)(lds_byte*)p;
}

// ---------------------------------------------------------------------------
// Tensor Data Mover: 2-D tile load, global -> LDS (CDNA5 ISA-8).
// D# group0: count=1 | lds_addr | global_addr | type=2
// D# group1: data_size=4B, pad_enable, pad_interval/amount codes,
//            tensor_dim0/1, tile_dim0/1, tensor_dim0_stride.
// Groups 2/3 zero (2-D tensor); trailing i32x8 arg of the 6-arg builtin is
// likewise zero. Completion tracked with TENSORcnt.
// pad_interval code: interval = 2^(code+1) dwords; pad_amount code: (code+1)
// dwords inserted -> reproduces padded LDS row strides for bank-conflict-free
// WMMA fragment reads.
// ---------------------------------------------------------------------------
static __device__ __forceinline__ void tdm_load_2d(
    unsigned lds_addr, const void* gaddr,
    unsigned tensor_d0, unsigned tensor_d1, unsigned long long stride0,
    unsigned tile_d0, unsigned tile_d1,
    unsigned pad_interval, unsigned pad_amount)
{
    const unsigned long long ga = (unsigned long long)(uintptr_t)gaddr;
    u32x4 g0;
    g0[0] = 1u;                                    // count=1, user mode
    g0[1] = lds_addr;                              // LDS byte address
    g0[2] = (unsigned)ga;                          // global_addr[31:0]
    g0[3] = (unsigned)((ga >> 32) & 0x01FFFFFFu)   // global_addr[56:32]
          | 0x80000000u;                           // type=2 ("image")
    i32x8 g1;
    g1[0] = (int)((2u << 16)                       // data_size = 4 bytes
                | (1u << 20)                       // pad_enable
                | (pad_interval << 22)
                | (pad_amount << 25));
    g1[1] = (int)((tensor_d0 & 0xFFFFu) << 16);    // abar_addr=0 | td0[15:0]
    g1[2] = (int)(((tensor_d0 >> 16) & 0xFFFFu) | ((tensor_d1 & 0xFFFFu) << 16));
    g1[3] = (int)(((tensor_d1 >> 16) & 0xFFFFu) | ((tile_d0 & 0xFFFFu) << 16));
    g1[4] = (int)(tile_d1 & 0xFFFFu);              // tile_dim2 = 0
    g1[5] = (int)(unsigned)(stride0 & 0xFFFFFFFFull);
    g1[6] = (int)(unsigned)((stride0 >> 32) & 0xFFFFull); // dim1_stride = 0
    g1[7] = 0;
    i32x4 g2 = {0, 0, 0, 0};
    i32x4 g3 = {0, 0, 0, 0};
    i32x8 g4 = {0, 0, 0, 0, 0, 0, 0, 0};
    __builtin_amdgcn_tensor_load_to_lds(g0, g1, g2, g3, g4, 0);
}

// ------------------------------ Phase 1 ------------------------------------
// 256 threads = 8 waves. Macro tile 64(M) x 32(N); wave (msub,nsub) owns a
// 16x16 WMMA tile. K in 64-wide chunks, TDM double-buffered into LDS with
// row stride 68 floats (pad 4 dwords every 64 -> 4s mod 64 bank map is
// conflict-free for fragment reads where lane%16 selects the row).
__global__ __launch_bounds__(256) void wx_gemm_kernel(
    const float* __restrict__ x, const float* __restrict__ W,
    const float* __restrict__ Ub, const float* __restrict__ bb,
    float* __restrict__ out)
{
    __shared__ float xs[2][64 * 68];   // x chunk  [64 m][64 k] + pad
    __shared__ float ws[2][32 * 68];   // W chunk  [32 n][64 k] + pad

    const int tid  = threadIdx.x;
    const int wid  = tid >> 5;      // wave32
    const int lane = tid & 31;
    const int lm   = lane & 15;
    const int lh   = lane >> 4;

    const int mb = blockIdx.x >> 5;        // MTOT/64 = 2048 m-blocks
    const int nb = blockIdx.x & 31;        // HH/32   = 32 n-blocks
    const int m0 = mb * 64;
    const int n0 = nb * 32;

    const int msub = wid >> 1;             // 0..3
    const int nsub = wid & 1;              // 0..1

    v8f c = {};

    // Prologue: DMA chunk 0 into buffer 0 (pad: 4 dwords every 64 -> stride 68)
    if (wid == 0) {
        tdm_load_2d(lds_off(&xs[0][0]), &x[(size_t)m0 * KIN],
                    KIN, MTOT, KIN, 64, 64, /*interval=*/5, /*amount=*/3);
        tdm_load_2d(lds_off(&ws[0][0]), &W[(size_t)n0 * KIN],
                    KIN, HH, KIN, 64, 32, 5, 3);
    }

    for (int kci = 0; kci < KIN / 64; ++kci) {
        const int cur = kci & 1;
        if (wid == 0) {
            if (kci + 1 < KIN / 64) {
                const int kc = (kci + 1) * 64;
                // issue next chunk's DMAs, then wait only for current chunk
                tdm_load_2d(lds_off(&xs[cur ^ 1][0]),
                            &x[(size_t)m0 * KIN + kc], KIN, MTOT, KIN, 64, 64, 5, 3);
                tdm_load_2d(lds_off(&ws[cur ^ 1][0]),
                            &W[(size_t)n0 * KIN + kc], KIN, HH, KIN, 64, 32, 5, 3);
                __builtin_amdgcn_s_wait_tensorcnt(2);
            } else {
                __builtin_amdgcn_s_wait_tensorcnt(0);
            }
        }
        __syncthreads();

        // A fragment: lane -> M = lane%16, K = kk + 2*(lane/16) + {0,1}
        // B fragment: lane -> N = lane%16, K = kk + 2*(lane/16) + {0,1}
        const float* ax = &xs[cur][(msub * 16 + lm) * 68 + 2 * lh];
        const float* bw = &ws[cur][(nsub * 16 + lm) * 68 + 2 * lh];
        #pragma unroll
        for (int kk = 0; kk < 64; kk += 4) {
            v2f a = *(const v2f*)(ax + kk);
            v2f b = *(const v2f*)(bw + kk);
            c = __builtin_amdgcn_wmma_f32_16x16x4_f32(
                    false, a, false, b, (short)0, c, false, false);
        }
        __syncthreads();
    }

    const int   n  = n0 + nsub * 16 + lm;
    const float ub = Ub[n] + bb[n];
    #pragma unroll
    for (int v = 0; v < 8; ++v) {
        // C layout: VGPR v -> M = v + 8*(lane/16), N = lane%16
        int m = m0 + msub * 16 + v + 8 * lh;
        out[(size_t)m * HH + n] = c[v] + ub;
    }
}

// ------------------------------ Phase 2 ------------------------------------
// 128 WGs of 64 threads (2 waves). WG = (bg, ng): batch rows bg*16..+15,
// columns ng*32..+31. Uw slice (32 x 1024 f32) is TDM-loaded once into LDS
// with 1 pad dword every 256 (row stride 1028 -> conflict-free; float index
// k maps to k + k/256). Per step the 32 WGs of a batch-group sync via a
// monotonic agent-scope counter; state rows of t-1 are read from d_out (L2).
__global__ __launch_bounds__(64) void rnn_scan_kernel(
    const float* __restrict__ Uw, float* __restrict__ out,
    unsigned* __restrict__ counters)
{
    extern __shared__ float uw[];   // [32][1028] floats (TDM pad layout)

    const int tid  = threadIdx.x;
    const int bg   = blockIdx.x >> 5;   // 0..3  batch group
    const int ng   = blockIdx.x & 31;   // 0..31 column group
    const int ncol0 = ng * 32;
    const int w    = tid >> 5;          // wave 0..1 -> 16-col subtile
    const int lane = tid & 31;
    const int lm   = lane & 15;
    const int lh   = lane >> 4;

    // Preload Uw rows [ncol0, ncol0+32): B[k][n] = Uw[n][k], row-major so the
    // B-fragment reads are contiguous float2 along k. Single TDM descriptor.
    if (w == 0) {
        tdm_load_2d(lds_off(uw), &Uw[(size_t)ncol0 * HH],
                    HH, HH, HH, /*tile_d0=*/1024, /*tile_d1=*/32,
                    /*interval=*/7, /*amount=*/0);   // 1 dword pad per 256
        __builtin_amdgcn_s_wait_tensorcnt(0);
    }
    __syncthreads();

    const int    colglob  = ncol0 + w * 16 + lm;          // this lane's N
    const size_t aRowBase = (size_t)(bg * 16 + lm) * THH; // this lane's A row
    const int    bBase    = (w * 16 + lm) * 1028 + 2 * lh;
    unsigned*    cnt      = counters + bg;

    for (int t = 0; t < SEQ; ++t) {
        const size_t tOff = (size_t)t * HH;

        // C <- wx[t] tile (C layout: VGPR v -> M = v + 8*lh, N = lane%16)
        v8f c;
        #pragma unroll
        for (int v = 0; v < 8; ++v)
            c[v] = out[(size_t)(bg * 16 + v + 8 * lh) * THH + tOff + colglob];

        if (t > 0) {
            const float* aPtr = out + aRowBase + (size_t)(t - 1) * HH + 2 * lh;
            const float* bPtr = uw + bBase;
            #pragma unroll 8
            for (int k0 = 0; k0 < HH; k0 += 4) {
                const int ko = k0 + (k0 >> 8);      // TDM pad interleave
                v2f a = *(const v2f*)(aPtr + k0);   // state[t-1] fragment
                v2f b = *(const v2f*)(bPtr + ko);   // Uw fragment from LDS
                c = __builtin_amdgcn_wmma_f32_16x16x4_f32(
                        false, a, false, b, (short)0, c, false, false);
            }
        }

        // store state[t] (in place over wx[t])
        #pragma unroll
        for (int v = 0; v < 8; ++v)
            out[(size_t)(bg * 16 + v + 8 * lh) * THH + tOff + colglob] = c[v];

        if (t + 1 < SEQ) {
            // warm next step's wx tile (global_prefetch_b8)
            __builtin_prefetch(out + aRowBase + tOff + HH + colglob, 0, 1);
            __threadfence();
            __syncthreads();
            if (tid == 0) {
                __hip_atomic_fetch_add(cnt, 1u, __ATOMIC_RELEASE,
                                       __HIP_MEMORY_SCOPE_AGENT);
                const unsigned target = (unsigned)(t + 1) * 32u;
                while (__hip_atomic_load(cnt, __ATOMIC_ACQUIRE,
                                         __HIP_MEMORY_SCOPE_AGENT) < target) {
                    __builtin_amdgcn_s_sleep(1);
                }
            }
            __syncthreads();
            __threadfence();   // acquire: don't read stale state[t] lines
        }
    }
}

__global__ void init_counters_kernel(unsigned* c)
{
    if (threadIdx.x < 4) c[threadIdx.x] = 0u;
}

// ------------------------------ launcher -----------------------------------
extern "C" void kernel_launch(void* const* d_in, const int* in_sizes, int n_in,
                              void* d_out, int out_size, void* d_ws, size_t ws_size,
                              hipStream_t stream)
{
    const float* x  = (const float*)d_in[0];   // [64, 2048, 512]
    const float* W  = (const float*)d_in[1];   // [1024, 512]
    const float* Uw = (const float*)d_in[2];   // [1024, 1024]
    const float* Ub = (const float*)d_in[3];   // [1024]
    const float* bb = (const float*)d_in[4];   // [1, 1024]
    float* out = (float*)d_out;                // [64, 2048, 1024]
    unsigned* counters = (unsigned*)d_ws;      // 4 per-batch-group barriers

    init_counters_kernel<<<dim3(1), dim3(64), 0, stream>>>(counters);

    // Phase 1: (131072/64) * (1024/32) = 65536 blocks
    wx_gemm_kernel<<<dim3(65536), dim3(256), 0, stream>>>(x, W, Ub, bb, out);

    // Phase 2: 4 batch groups x 32 column groups, 128.5 KB dynamic LDS each
    rnn_scan_kernel<<<dim3(128), dim3(64), 32 * 1028 * sizeof(float), stream>>>(
        Uw, out, counters);
}